// LowRankGMM_25967372272263
// MI455X (gfx1250) — compile-verified
//
#include <hip/hip_runtime.h>
#include <math.h>

#define DIM      64
#define RANK     8
#define KCOMP    16
#define SPLIT_SUM 641            // 1 + 64 + 64*8 + 64
#define OUTD     (SPLIT_SUM * KCOMP)  // 10256
#define O_MU     1
#define O_W      65
#define O_LD     577
#define LOG2PI   1.8378770664093453f

typedef float v2f __attribute__((ext_vector_type(2)));
typedef float v8f __attribute__((ext_vector_type(8)));

__global__ __launch_bounds__(256) void gmm_nll_kernel(
    const float* __restrict__ pred,
    const float* __restrict__ label,
    float* __restrict__ out)
{
    __shared__ float sRow[OUTD];               // full pred row (pi,mu,W,logd)*16
    __shared__ float sDinv[KCOMP][DIM];        // exp(-logd)
    __shared__ float sCap[KCOMP][RANK][RANK];  // [k][col][row], I + W^T D^-1 W
    __shared__ float sLabel[DIM];
    __shared__ float sLp[KCOMP];
    __shared__ float sPi[KCOMP];

    const int b   = blockIdx.x;
    const int tid = threadIdx.x;
    const float* row = pred + (size_t)b * OUTD;

    // ---- stage pred row + label into LDS (coalesced b32) ----
    for (int i = tid; i < OUTD; i += 256) sRow[i] = row[i];
    if (tid < DIM) sLabel[tid] = label[(size_t)b * DIM + tid];
    __syncthreads();

    // ---- dinv = exp(-logd) ----
    for (int i = tid; i < KCOMP * DIM; i += 256) {
        int k = i >> 6, d = i & 63;
        sDinv[k][d] = expf(-sRow[k * SPLIT_SUM + O_LD + d]);
    }
    __syncthreads();

    // ---- per-wave: two components, cap via V_WMMA_F32_16X16X4_F32 ----
    const int wave = tid >> 5;
    const int lane = tid & 31;
    const int hi   = lane >> 4;   // 0: lanes 0-15 (K=0,1), 1: lanes 16-31 (K=2,3)
    const int m    = lane & 15;   // A-row index == B-col index for this lane
    const int k0   = wave * 2;
    const int kAB  = k0 + (m >> 3);      // component for this lane's row/col
    const int rAB  = m & 7;              // rank index within component
    const int wBase = kAB * SPLIT_SUM + O_W + rAB;  // W[k][d][r] = sRow[wBase + d*8]

    v8f acc = {0.f, 0.f, 0.f, 0.f, 0.f, 0.f, 0.f, 0.f};
#pragma unroll
    for (int it = 0; it < 16; ++it) {
        const int d0 = it * 4 + (hi ? 2 : 0);
        const float w0 = sRow[wBase + d0 * RANK];
        const float w1 = sRow[wBase + (d0 + 1) * RANK];
        v2f Af = { w0 * sDinv[kAB][d0], w1 * sDinv[kAB][d0 + 1] };  // scaled W^T
        v2f Bf = { w0, w1 };                                         // raw W
        acc = __builtin_amdgcn_wmma_f32_16x16x4_f32(
            false, Af, false, Bf, (short)0, acc, false, false);
    }

    // D layout: VGPR v, lane L -> M = v + (L>=16 ? 8 : 0), N = L & 15.
    // comp k0 block: M,N in 0..7  -> lanes 0..7.   comp k1 block: M,N in 8..15 -> lanes 24..31.
    if (hi == 0 && m < 8) {
#pragma unroll
        for (int v = 0; v < 8; ++v)
            sCap[k0][m][v] = acc[v] + (v == m ? 1.0f : 0.0f);
    }
    if (hi == 1 && m >= 8) {
        const int col = m - 8;
#pragma unroll
        for (int v = 0; v < 8; ++v)
            sCap[k0 + 1][col][v] = acc[v] + (v == col ? 1.0f : 0.0f);
    }

    // ---- scalar path: each 16-lane half owns one component; lane owns 4 d's ----
    const int k    = k0 + hi;
    const int base = k * SPLIT_SUM;
    float mah = 0.f, sld = 0.f;
    float bv[8] = {0.f, 0.f, 0.f, 0.f, 0.f, 0.f, 0.f, 0.f};
#pragma unroll
    for (int j = 0; j < 4; ++j) {
        const int d = m * 4 + j;
        const float diff = sLabel[d] - sRow[base + O_MU + d];
        const float ddv  = diff * sDinv[k][d];
        mah += diff * ddv;
        sld += sRow[base + O_LD + d];
#pragma unroll
        for (int r = 0; r < 8; ++r)
            bv[r] += sRow[base + O_W + d * RANK + r] * ddv;
    }
    // reduce across the 16-lane half (xor masks 1,2,4,8 stay within the half)
#pragma unroll
    for (int mask = 1; mask <= 8; mask <<= 1) {
        mah += __shfl_xor(mah, mask, 32);
        sld += __shfl_xor(sld, mask, 32);
#pragma unroll
        for (int r = 0; r < 8; ++r) bv[r] += __shfl_xor(bv[r], mask, 32);
    }
    __syncthreads();  // make sCap visible

    // ---- Cholesky + solve: lane 0 of each half handles its component ----
    if (m == 0) {
        float A_[36];  // lower triangle, tri(i,j) = i*(i+1)/2 + j
#pragma unroll
        for (int i = 0; i < 8; ++i)
#pragma unroll
            for (int j = 0; j <= i; ++j)
                A_[i * (i + 1) / 2 + j] = sCap[k][j][i];  // cap[i][j]

        float slogd = 0.f;
#pragma unroll
        for (int j = 0; j < 8; ++j) {
            float s = A_[j * (j + 1) / 2 + j];
#pragma unroll
            for (int t = 0; t < j; ++t) {
                const float l = A_[j * (j + 1) / 2 + t];
                s -= l * l;
            }
            const float dj = sqrtf(s);
            A_[j * (j + 1) / 2 + j] = dj;
            slogd += logf(dj);
            const float inv = 1.0f / dj;
#pragma unroll
            for (int i = j + 1; i < 8; ++i) {
                float v = A_[i * (i + 1) / 2 + j];
#pragma unroll
                for (int t = 0; t < j; ++t)
                    v -= A_[i * (i + 1) / 2 + t] * A_[j * (j + 1) / 2 + t];
                A_[i * (i + 1) / 2 + j] = v * inv;
            }
        }
        // forward substitution L y = bv
        float y[8];
        float sy2 = 0.f;
#pragma unroll
        for (int i = 0; i < 8; ++i) {
            float v = bv[i];
#pragma unroll
            for (int t = 0; t < i; ++t)
                v -= A_[i * (i + 1) / 2 + t] * y[t];
            y[i] = v / A_[i * (i + 1) / 2 + i];
            sy2 += y[i] * y[i];
        }
        const float mahal  = mah - sy2;
        const float logdet = 2.f * slogd + sld;
        sLp[k] = -0.5f * ((float)DIM * LOG2PI + logdet + mahal);
        sPi[k] = sRow[base];  // raw mixture logit
    }
    __syncthreads();

    // ---- final logsumexp over 16 components (thread 0) ----
    if (tid == 0) {
        float t[KCOMP];
        float mp = -INFINITY, mt = -INFINITY;
#pragma unroll
        for (int kk = 0; kk < KCOMP; ++kk) {
            const float p = sPi[kk];
            t[kk] = p + sLp[kk];
            mp = fmaxf(mp, p);
            mt = fmaxf(mt, t[kk]);
        }
        float sp = 0.f, st = 0.f;
#pragma unroll
        for (int kk = 0; kk < KCOMP; ++kk) {
            sp += expf(sPi[kk] - mp);
            st += expf(t[kk] - mt);
        }
        const float lse_pi = logf(sp) + mp;  // logsumexp(pis)
        const float lse_t  = logf(st) + mt;  // logsumexp(pis + comp_lp)
        out[b] = -(lse_t - lse_pi);
    }
}

extern "C" void kernel_launch(void* const* d_in, const int* in_sizes, int n_in,
                              void* d_out, int out_size, void* d_ws, size_t ws_size,
                              hipStream_t stream) {
    const float* pred  = (const float*)d_in[0];
    const float* lbl   = (const float*)d_in[1];
    float*       outp  = (float*)d_out;
    const int B = in_sizes[0] / OUTD;  // 8192
    gmm_nll_kernel<<<B, 256, 0, stream>>>(pred, lbl, outp);
}